// AttModel_7086696038514
// MI455X (gfx1250) — compile-verified
//
#include <hip/hip_runtime.h>
#include <hip/hip_bf16.h>

typedef float v2f __attribute__((ext_vector_type(2)));
typedef float v8f __attribute__((ext_vector_type(8)));

#define BB     32
#define NN     512
#define INF_   128
#define HH     8
#define DD     16
#define OUTF   128
#define TOPK_  16
#define NEGV   (-1.0e9f)
#define SCALE_ 0.25f   // 1/sqrt(16)

// ---------------------------------------------------------------------------
// Kernel 1: fused QKV projection.  Y = relu(X @ W + b)
// X: [B*N, 128] row-major, W: [128,128], out layout: [B, H, N, 16]
// grid: (B*N/16, 3), block: 256 (8 waves; wave w -> cols 16w..16w+15)
// ---------------------------------------------------------------------------
__global__ __launch_bounds__(256) void qkv_proj_kernel(
    const float* __restrict__ x,
    const float* __restrict__ Wq, const float* __restrict__ bq,
    const float* __restrict__ Wk, const float* __restrict__ bk,
    const float* __restrict__ Wv, const float* __restrict__ bv,
    float* __restrict__ qo, float* __restrict__ ko, float* __restrict__ vo)
{
    const int tile  = blockIdx.x;          // 16-row tile of X (never crosses batch: 512%16==0)
    const int which = blockIdx.y;          // 0=q 1=k 2=v
    const int wave  = threadIdx.x >> 5;
    const int lane  = threadIdx.x & 31;
    const int half  = lane >> 4;
    const int l     = lane & 15;

    const float* W;  const float* bias;  float* out;
    if (which == 0)      { W = Wq; bias = bq; out = qo; }
    else if (which == 1) { W = Wk; bias = bk; out = ko; }
    else                 { W = Wv; bias = bv; out = vo; }

    const int row0 = tile * 16;
    const int col0 = wave * 16;

    const float* xrow = x + (size_t)(row0 + l) * INF_;

    v8f acc = {};
#pragma unroll
    for (int k0 = 0; k0 < INF_; k0 += 4) {
        const int ka = k0 + 2 * half;
        v2f a;  a.x = xrow[ka];                    a.y = xrow[ka + 1];
        v2f bf; bf.x = W[ka * OUTF + col0 + l];    bf.y = W[(ka + 1) * OUTF + col0 + l];
        acc = __builtin_amdgcn_wmma_f32_16x16x4_f32(false, a, false, bf,
                                                    (short)0, acc, false, false);
    }

    const int col = col0 + l;
    const float bb = bias[col];
    const int h = col >> 4, d = col & 15;
#pragma unroll
    for (int i = 0; i < 8; ++i) {
        const int m = i + 8 * half;
        const int r = row0 + m;
        const int bidx = r >> 9;        // / 512
        const int n    = r & 511;
        float val = acc[i] + bb;
        val = val > 0.f ? val : 0.f;
        out[(((size_t)bidx * HH + h) * NN + n) * DD + d] = val;
    }
}

// ---------------------------------------------------------------------------
// Kernel 2: attention core.  Per block: (b, h, 16-query tile).
// Phase 1 (WMMA): logits = Q @ K^T * SCALE  -> stream to att_logits (d_out),
//                 masked copy into LDS.
// Phase 2: exact top-16 per row (wave-wide argmax x16), softmax over the 16
//          survivors, sparse P@V + v residual -> o_heads [B, N, H*16].
// grid: (N/16, H, B), block: 256
// ---------------------------------------------------------------------------
__global__ __launch_bounds__(256) void attn_kernel(
    const float* __restrict__ qg, const float* __restrict__ kg,
    const float* __restrict__ vg, const int* __restrict__ mask,
    float* __restrict__ att_out, float* __restrict__ o_heads)
{
    __shared__ float sm[16 * 516];      // masked logits, padded stride
    __shared__ float s_val[16 * 16];    // top-16 values per row
    __shared__ int   s_idx[16 * 16];    // top-16 indices per row

    const int qt   = blockIdx.x;
    const int h    = blockIdx.y;
    const int b    = blockIdx.z;
    const int wave = threadIdx.x >> 5;
    const int lane = threadIdx.x & 31;
    const int half = lane >> 4;
    const int l    = lane & 15;
    const int q0   = qt * 16;

    const size_t bh = (size_t)b * HH + h;
    const float* Qb = qg + bh * NN * DD;
    const float* Kb = kg + bh * NN * DD;
    const float* Vb = vg + bh * NN * DD;

    // Preload Q fragments (shared across all K-tiles of this wave).
    v2f aq[4];
    {
        const float* qrow = Qb + (size_t)(q0 + l) * DD;
#pragma unroll
        for (int s = 0; s < 4; ++s) {
            const int kd = s * 4 + 2 * half;
            aq[s].x = qrow[kd];  aq[s].y = qrow[kd + 1];
        }
    }

    float*     attb  = att_out + (bh * NN + q0) * (size_t)NN;
    const int* maskb = mask + ((size_t)b * NN + q0) * NN;

    // ---- Phase 1: QK^T tiles, stream logits, build masked LDS copy ----
#pragma unroll
    for (int j = 0; j < 4; ++j) {
        const int k0c = (wave * 4 + j) * 16;
        const float* krow = Kb + (size_t)(k0c + l) * DD;
        v8f acc = {};
#pragma unroll
        for (int s = 0; s < 4; ++s) {
            const int kd = s * 4 + 2 * half;
            v2f bf; bf.x = krow[kd]; bf.y = krow[kd + 1];
            acc = __builtin_amdgcn_wmma_f32_16x16x4_f32(false, aq[s], false, bf,
                                                        (short)0, acc, false, false);
        }
        const int kc = k0c + l;
#pragma unroll
        for (int i = 0; i < 8; ++i) {
            const int m = i + 8 * half;
            const float logit = acc[i] * SCALE_;
            attb[(size_t)m * NN + kc] = logit;                      // att_logits output
            const int mk = maskb[m * NN + kc];
            sm[m * 516 + kc] = (mk != 0) ? logit : NEGV;
        }
    }
    __syncthreads();

    // ---- Phase 2: top-16 + softmax + sparse PV (2 rows per wave) ----
    for (int r = 0; r < 2; ++r) {
        const int row = wave * 2 + r;

        float lv[16];
#pragma unroll
        for (int t = 0; t < 16; ++t) lv[t] = sm[row * 516 + t * 32 + lane];

#pragma unroll
        for (int pass = 0; pass < 16; ++pass) {
            float lmax = -3.0e38f;
            int   lidx = 0x7fffffff;
#pragma unroll
            for (int t = 0; t < 16; ++t) {
                if (lv[t] > lmax) { lmax = lv[t]; lidx = t * 32 + lane; }
            }
            // wave-wide argmax (symmetric xor reduction: every lane converges)
#pragma unroll
            for (int off = 16; off >= 1; off >>= 1) {
                const float ov = __shfl_xor(lmax, off, 32);
                const int   oi = __shfl_xor(lidx, off, 32);
                if (ov > lmax || (ov == lmax && oi < lidx)) { lmax = ov; lidx = oi; }
            }
            if (lane == 0) { s_val[row * 16 + pass] = lmax; s_idx[row * 16 + pass] = lidx; }
            // retire exactly the winning element
#pragma unroll
            for (int t = 0; t < 16; ++t)
                if (t * 32 + lane == lidx) lv[t] = -3.0e38f;
        }

        // softmax over the 16 selected entries (the other 496 are -1e9 -> exp==0)
        const float rmax = s_val[row * 16 + 0];
        float ssum = 0.f;
#pragma unroll
        for (int i = 0; i < 16; ++i) ssum += __expf(s_val[row * 16 + i] - rmax);
        const float inv = 1.f / ssum;

        // sparse P@V: lanes 0-15 handle i=0..7, lanes 16-31 handle i=8..15
        float acc = 0.f;
#pragma unroll
        for (int t = 0; t < 8; ++t) {
            const int i = half * 8 + t;
            const float p = __expf(s_val[row * 16 + i] - rmax) * inv;
            acc += p * Vb[(size_t)s_idx[row * 16 + i] * DD + l];
        }
        acc += __shfl_xor(acc, 16, 32);

        if (half == 0) {
            const int qrow = q0 + row;
            const float res = acc + Vb[(size_t)qrow * DD + l];      // residual +v
            o_heads[((size_t)b * NN + qrow) * (HH * DD) + h * DD + l] = res;
        }
    }
}

// ---------------------------------------------------------------------------
// Kernel 3: output projection.  out = relu(Oh @ Wout + bout), Oh: [B*N,128]
// grid: (B*N/16), block: 256
// ---------------------------------------------------------------------------
__global__ __launch_bounds__(256) void out_proj_kernel(
    const float* __restrict__ oh, const float* __restrict__ Wout,
    const float* __restrict__ bout, float* __restrict__ out)
{
    const int tile = blockIdx.x;
    const int wave = threadIdx.x >> 5;
    const int lane = threadIdx.x & 31;
    const int half = lane >> 4;
    const int l    = lane & 15;

    const int row0 = tile * 16;
    const int col0 = wave * 16;
    const float* xrow = oh + (size_t)(row0 + l) * (HH * DD);

    v8f acc = {};
#pragma unroll
    for (int k0 = 0; k0 < HH * DD; k0 += 4) {
        const int ka = k0 + 2 * half;
        v2f a;  a.x = xrow[ka];                     a.y = xrow[ka + 1];
        v2f bf; bf.x = Wout[ka * OUTF + col0 + l];  bf.y = Wout[(ka + 1) * OUTF + col0 + l];
        acc = __builtin_amdgcn_wmma_f32_16x16x4_f32(false, a, false, bf,
                                                    (short)0, acc, false, false);
    }

    const int col = col0 + l;
    const float bb = bout[col];
#pragma unroll
    for (int i = 0; i < 8; ++i) {
        const int m = i + 8 * half;
        float val = acc[i] + bb;
        val = val > 0.f ? val : 0.f;
        out[(size_t)(row0 + m) * OUTF + col] = val;
    }
}

// ---------------------------------------------------------------------------
extern "C" void kernel_launch(void* const* d_in, const int* in_sizes, int n_in,
                              void* d_out, int out_size, void* d_ws, size_t ws_size,
                              hipStream_t stream)
{
    const float* x    = (const float*)d_in[0];
    const int*   mask = (const int*)  d_in[1];
    const float* Wq   = (const float*)d_in[2];
    const float* bq   = (const float*)d_in[3];
    const float* Wk   = (const float*)d_in[4];
    const float* bk   = (const float*)d_in[5];
    const float* Wv   = (const float*)d_in[6];
    const float* bv   = (const float*)d_in[7];
    const float* Wout = (const float*)d_in[8];
    const float* bout = (const float*)d_in[9];

    float* out        = (float*)d_out;                          // [B,N,128]
    float* att_logits = out + (size_t)BB * NN * OUTF;           // [B,H,N,N]

    const size_t per = (size_t)BB * HH * NN * DD;               // 2,097,152 elems
    float* qws = (float*)d_ws;
    float* kws = qws + per;
    float* vws = kws + per;
    float* ohe = vws + per;                                     // [B,N,H*16]

    // 1) fused QKV projection (q/k/v selected by grid.y)
    {
        dim3 grid(BB * NN / 16, 3);
        qkv_proj_kernel<<<grid, 256, 0, stream>>>(x, Wq, bq, Wk, bk, Wv, bv,
                                                  qws, kws, vws);
    }
    // 2) attention: logits stream + top-k softmax + sparse PV
    {
        dim3 grid(NN / 16, HH, BB);
        attn_kernel<<<grid, 256, 0, stream>>>(qws, kws, vws, mask,
                                              att_logits, ohe);
    }
    // 3) output projection
    {
        dim3 grid(BB * NN / 16);
        out_proj_kernel<<<grid, 256, 0, stream>>>(ohe, Wout, bout, out);
    }
}